// EAGRModule_47459388621085
// MI455X (gfx1250) — compile-verified
//
#include <hip/hip_runtime.h>
#include <cstdint>
#include <cstddef>

typedef __attribute__((ext_vector_type(2))) float v2f;
typedef __attribute__((ext_vector_type(8))) float v8f;

#define ABN_EPS  1e-5f
#define ABN_LEAK 0.01f

enum { EPI_NONE = 0, EPI_ABN = 1, EPI_RELU = 3, EPI_COLSCALE = 4 };

__device__ __forceinline__ float leaky(float y) {
    return (y >= 0.0f) ? y : ABN_LEAK * y;
}

// ---------------------------------------------------------------------------
// fp32 WMMA GEMM, register-blocked:  C[M,N] (+)= A[M,K] (row stride lda) * B[K,N]
// block = 128 threads = 4 waves; each wave computes a 32x32 macro-tile
// (2 m-tiles x 2 n-tiles) -> per k-step: 2 A-frag + 2 B-frag loads, 4 WMMAs.
// Waves of a block cover 128 consecutive N columns. K must be a multiple of 4.
// Lane mapping per CDNA5 ISA 7.12.2 for V_WMMA_F32_16X16X4_F32.
// ---------------------------------------------------------------------------
template<int MODE, bool ACCUM>
__global__ __launch_bounds__(128) void gemm_wmma_f32(
    const float* __restrict__ A, size_t sAb, int lda,
    const float* __restrict__ B, size_t sBb,
    float* __restrict__ C, size_t sCb,
    int M, int N, int K,
    const float* __restrict__ q0, const float* __restrict__ q1,
    const float* __restrict__ q2, const float* __restrict__ q3, size_t sQb)
{
    const int wave = threadIdx.x >> 5;
    const int lane = threadIdx.x & 31;
    const int half = lane >> 4;
    const int l16  = lane & 15;
    const int m0 = blockIdx.y * 32;
    const int m1 = m0 + 16;
    const int n0 = blockIdx.x * 128 + wave * 32;
    const int n1 = n0 + 16;
    const int batch = blockIdx.z;
    if (n0 >= N || m0 >= M) return;     // wave-uniform

    A += (size_t)batch * sAb;
    B += (size_t)batch * sBb;
    C += (size_t)batch * sCb;

    const bool c1ok = (n1 < N);         // second n-tile exists (wave-uniform, N % 16 == 0)
    const int  arow0 = m0 + l16;
    const int  arow1 = m1 + l16;
    const bool rOK0 = arow0 < M;
    const bool rOK1 = arow1 < M;
    const int  col0 = n0 + l16;
    const int  col1 = n1 + l16;
    const float* Ar0 = A + (size_t)arow0 * lda;
    const float* Ar1 = A + (size_t)arow1 * lda;

    v8f acc00, acc01, acc10, acc11;
    #pragma unroll
    for (int r = 0; r < 8; ++r) {
        if (ACCUM) {
            int rr0 = m0 + r + 8 * half;
            int rr1 = m1 + r + 8 * half;
            acc00[r] = (rr0 < M)          ? C[(size_t)rr0 * N + col0] : 0.0f;
            acc01[r] = (rr0 < M && c1ok)  ? C[(size_t)rr0 * N + col1] : 0.0f;
            acc10[r] = (rr1 < M)          ? C[(size_t)rr1 * N + col0] : 0.0f;
            acc11[r] = (rr1 < M && c1ok)  ? C[(size_t)rr1 * N + col1] : 0.0f;
        } else {
            acc00[r] = 0.0f; acc01[r] = 0.0f; acc10[r] = 0.0f; acc11[r] = 0.0f;
        }
    }

    #pragma unroll 2
    for (int k = 0; k < K; k += 4) {
        const int kk = k + 2 * half;
        v2f a0, a1, b0, b1;
        a0.x = rOK0 ? Ar0[kk]     : 0.0f;
        a0.y = rOK0 ? Ar0[kk + 1] : 0.0f;
        a1.x = rOK1 ? Ar1[kk]     : 0.0f;
        a1.y = rOK1 ? Ar1[kk + 1] : 0.0f;
        b0.x = B[(size_t)kk * N + col0];
        b0.y = B[(size_t)(kk + 1) * N + col0];
        b1.x = c1ok ? B[(size_t)kk * N + col1]       : 0.0f;
        b1.y = c1ok ? B[(size_t)(kk + 1) * N + col1] : 0.0f;
        acc00 = __builtin_amdgcn_wmma_f32_16x16x4_f32(false, a0, false, b0, (short)0, acc00, false, false);
        acc01 = __builtin_amdgcn_wmma_f32_16x16x4_f32(false, a0, false, b1, (short)0, acc01, false, false);
        acc10 = __builtin_amdgcn_wmma_f32_16x16x4_f32(false, a1, false, b0, (short)0, acc10, false, false);
        acc11 = __builtin_amdgcn_wmma_f32_16x16x4_f32(false, a1, false, b1, (short)0, acc11, false, false);
    }

    float cs0 = 1.0f, cs1 = 1.0f;
    if (MODE == EPI_COLSCALE) {
        cs0 = q0[(size_t)batch * sQb + col0];
        cs1 = c1ok ? q0[(size_t)batch * sQb + col1] : 0.0f;
    }

    #pragma unroll
    for (int r = 0; r < 8; ++r) {
        int rr0 = m0 + r + 8 * half;
        if (rr0 < M) {
            float y0 = acc00[r], y1 = acc01[r];
            if (MODE == EPI_ABN) {
                float sc = q0[rr0] * rsqrtf(q3[rr0] + ABN_EPS);
                float mm = q2[rr0], bb = q1[rr0];
                y0 = leaky((y0 - mm) * sc + bb);
                y1 = leaky((y1 - mm) * sc + bb);
            } else if (MODE == EPI_RELU) {
                y0 = fmaxf(y0, 0.0f); y1 = fmaxf(y1, 0.0f);
            } else if (MODE == EPI_COLSCALE) {
                y0 *= cs0; y1 *= cs1;
            }
            C[(size_t)rr0 * N + col0] = y0;
            if (c1ok) C[(size_t)rr0 * N + col1] = y1;
        }
        int rr1 = m1 + r + 8 * half;
        if (rr1 < M) {
            float y0 = acc10[r], y1 = acc11[r];
            if (MODE == EPI_ABN) {
                float sc = q0[rr1] * rsqrtf(q3[rr1] + ABN_EPS);
                float mm = q2[rr1], bb = q1[rr1];
                y0 = leaky((y0 - mm) * sc + bb);
                y1 = leaky((y1 - mm) * sc + bb);
            } else if (MODE == EPI_RELU) {
                y0 = fmaxf(y0, 0.0f); y1 = fmaxf(y1, 0.0f);
            } else if (MODE == EPI_COLSCALE) {
                y0 *= cs0; y1 *= cs1;
            }
            C[(size_t)rr1 * N + col0] = y0;
            if (c1ok) C[(size_t)rr1 * N + col1] = y1;
        }
    }
}

// ---------------------------------------------------------------------------
// e = softmax(nearest_up(edge))[ch 1] = sigmoid(e1 - e0), 4 x 16384
// ---------------------------------------------------------------------------
__global__ __launch_bounds__(256) void edge_softmax_kernel(
    const float* __restrict__ edge, float* __restrict__ e)
{
    int t = blockIdx.x * 256 + threadIdx.x;
    if (t >= 4 * 16384) return;
    int n = t >> 14;
    int p = t & 16383;
    int iy = (p >> 7) >> 1;
    int ix = (p & 127) >> 1;
    const float* eb = edge + (size_t)n * 2 * 4096;
    float e0 = eb[iy * 64 + ix];
    float e1 = eb[4096 + iy * 64 + ix];
    e[t] = 1.0f / (1.0f + expf(e0 - e1));
}

// ---------------------------------------------------------------------------
// x = bilinear_align_corners(u, 64->128) + cs_b[c];  (4,256,128,128)
// ---------------------------------------------------------------------------
__global__ __launch_bounds__(256) void upsample_bias_kernel(
    const float* __restrict__ u, const float* __restrict__ bias, float* __restrict__ x)
{
    int t = blockIdx.x * 256 + threadIdx.x;
    if (t >= 4 * 256 * 16384) return;
    int ox = t & 127;
    int oy = (t >> 7) & 127;
    int c  = (t >> 14) & 255;
    const float s = 63.0f / 127.0f;
    float fy = oy * s, fx = ox * s;
    int y0 = min((int)fy, 63), x0 = min((int)fx, 63);
    int y1 = min(y0 + 1, 63),  x1 = min(x0 + 1, 63);
    float wy = fy - (float)y0, wx = fx - (float)x0;
    const float* up = u + ((size_t)(t >> 14)) * 4096;  // (n*256+c)*4096
    float v00 = up[y0 * 64 + x0], v01 = up[y0 * 64 + x1];
    float v10 = up[y1 * 64 + x0], v11 = up[y1 * 64 + x1];
    float v = (v00 * (1.0f - wx) + v01 * wx) * (1.0f - wy)
            + (v10 * (1.0f - wx) + v11 * wx) * wy;
    x[t] = v + bias[c];
}

// ---------------------------------------------------------------------------
// top-4 per (n,c) row of x_low (16384 values); indices sorted by value desc,
// ties -> lower index first (jax.lax.top_k semantics).
// ---------------------------------------------------------------------------
__global__ __launch_bounds__(256) void topk4_kernel(
    const float* __restrict__ xlow, int* __restrict__ idxb)
{
    __shared__ float sv[1024];
    __shared__ int   si[1024];
    const int rowid = blockIdx.x;
    const float* row = xlow + (size_t)rowid * 16384;
    float bv[4] = {-3.4e38f, -3.4e38f, -3.4e38f, -3.4e38f};
    int   bi[4] = {0x7fffffff, 0x7fffffff, 0x7fffffff, 0x7fffffff};
    for (int p = threadIdx.x; p < 16384; p += 256) {
        float nv = row[p]; int ni = p;
        #pragma unroll
        for (int j = 0; j < 4; ++j) {
            bool better = (nv > bv[j]) || (nv == bv[j] && ni < bi[j]);
            if (better) { float tv = bv[j]; int ti = bi[j]; bv[j] = nv; bi[j] = ni; nv = tv; ni = ti; }
        }
    }
    #pragma unroll
    for (int j = 0; j < 4; ++j) { sv[threadIdx.x * 4 + j] = bv[j]; si[threadIdx.x * 4 + j] = bi[j]; }
    __syncthreads();
    if (threadIdx.x == 0) {
        float fv[4] = {-3.4e38f, -3.4e38f, -3.4e38f, -3.4e38f};
        int   fi[4] = {0x7fffffff, 0x7fffffff, 0x7fffffff, 0x7fffffff};
        for (int t = 0; t < 1024; ++t) {
            float nv = sv[t]; int ni = si[t];
            for (int j = 0; j < 4; ++j) {
                bool better = (nv > fv[j]) || (nv == fv[j] && ni < fi[j]);
                if (better) { float tv = fv[j]; int ti = fi[j]; fv[j] = nv; fi[j] = ni; nv = tv; ni = ti; }
            }
        }
        for (int j = 0; j < 4; ++j) idxb[rowid * 4 + j] = fi[j];
    }
}

// ---------------------------------------------------------------------------
// anchor_feature[n,k,s] = x[n,s,idx[n,k]] * (1 - e[n,idx[n,k]])
// ---------------------------------------------------------------------------
__global__ __launch_bounds__(256) void gather_anchors_kernel(
    const float* __restrict__ x, const float* __restrict__ e,
    const int* __restrict__ idxb, float* __restrict__ af)
{
    int t = blockIdx.x * 256 + threadIdx.x;
    if (t >= 4 * 80 * 256) return;
    int s = t & 255;
    int k = (t >> 8) % 80;
    int n = t / (80 * 256);
    int p = idxb[n * 80 + k];
    float val = x[((size_t)n * 256 + s) * 16384 + p] * (1.0f - e[n * 16384 + p]);
    af[((size_t)n * 80 + k) * 256 + s] = val;
}

// ---------------------------------------------------------------------------
// hg[n,s,i] = sum_j gw1[i,j]*af[n,j,s] + gb1[i] - af[n,i,s]   (store (n,s,i))
// ---------------------------------------------------------------------------
__global__ __launch_bounds__(256) void hg_kernel(
    const float* __restrict__ gw1, const float* __restrict__ gb1,
    const float* __restrict__ af, float* __restrict__ hg)
{
    int t = blockIdx.x * 256 + threadIdx.x;
    if (t >= 4 * 256 * 80) return;
    int i = t % 80;
    int s = (t / 80) % 256;
    int n = t / (80 * 256);
    const float* afn = af + (size_t)n * 80 * 256;
    float acc = gb1[i];
    for (int j = 0; j < 80; ++j)
        acc = fmaf(gw1[i * 80 + j], afn[j * 256 + s], acc);
    acc -= afn[i * 256 + s];
    hg[((size_t)n * 256 + s) * 80 + i] = acc;
}

// ---------------------------------------------------------------------------
// proj[:,p] = softmax_k(proj[:,p]) * e[p]   (proj already contains *e pre-softmax)
// ---------------------------------------------------------------------------
__global__ __launch_bounds__(256) void softmax_proj_kernel(
    float* __restrict__ proj, const float* __restrict__ e)
{
    int t = blockIdx.x * 256 + threadIdx.x;
    if (t >= 4 * 16384) return;
    int p = t & 16383, n = t >> 14;
    float* col = proj + (size_t)n * 80 * 16384 + p;
    float mx = -3.4e38f;
    for (int k = 0; k < 80; ++k) mx = fmaxf(mx, col[(size_t)k * 16384]);
    float s = 0.0f;
    for (int k = 0; k < 80; ++k) s += expf(col[(size_t)k * 16384] - mx);
    float scale = e[t] / s;
    for (int k = 0; k < 80; ++k) col[(size_t)k * 16384] = expf(col[(size_t)k * 16384] - mx) * scale;
}

// ---------------------------------------------------------------------------
// Fused: x3 = abn3(c3_w @ xf) into LDS (256 x 32 block), then
//        seg  = c4_w @ x3 + c4_b  (B read from LDS).  256 thr = 8 waves.
// Phase 1 register-blocked: each wave does 2 m-tiles x 2 n-tiles with
// shared A/B fragments (4 WMMAs per 4 loads).
// ---------------------------------------------------------------------------
__global__ __launch_bounds__(256) void fused_c3abn_c4_kernel(
    const float* __restrict__ X,
    const float* __restrict__ c3w, const float* __restrict__ g3, const float* __restrict__ b3,
    const float* __restrict__ m3, const float* __restrict__ v3,
    const float* __restrict__ c4w, const float* __restrict__ c4b,
    float* __restrict__ seg)
{
    __shared__ float lds[256 * 33];
    const int wave = threadIdx.x >> 5;
    const int lane = threadIdx.x & 31;
    const int half = lane >> 4;
    const int l16  = lane & 15;
    const int n    = blockIdx.y;
    const int col0 = blockIdx.x * 32;
    const float* Xb = X + (size_t)n * 256 * 16384;

    // phase 1: wave covers m rows [wave*32, wave*32+32), n cols [col0, col0+32)
    {
        const int mA0 = wave * 32;
        const int mA1 = mA0 + 16;
        const float* Ar0 = c3w + (size_t)(mA0 + l16) * 256;
        const float* Ar1 = c3w + (size_t)(mA1 + l16) * 256;
        const int cg0 = col0 + l16;
        const int cg1 = col0 + 16 + l16;

        v8f acc00, acc01, acc10, acc11;
        #pragma unroll
        for (int r = 0; r < 8; ++r) { acc00[r] = 0.0f; acc01[r] = 0.0f; acc10[r] = 0.0f; acc11[r] = 0.0f; }

        #pragma unroll 2
        for (int k = 0; k < 256; k += 4) {
            const int kk = k + 2 * half;
            v2f a0, a1, b0, b1;
            a0.x = Ar0[kk]; a0.y = Ar0[kk + 1];
            a1.x = Ar1[kk]; a1.y = Ar1[kk + 1];
            b0.x = Xb[(size_t)kk * 16384 + cg0];
            b0.y = Xb[(size_t)(kk + 1) * 16384 + cg0];
            b1.x = Xb[(size_t)kk * 16384 + cg1];
            b1.y = Xb[(size_t)(kk + 1) * 16384 + cg1];
            acc00 = __builtin_amdgcn_wmma_f32_16x16x4_f32(false, a0, false, b0, (short)0, acc00, false, false);
            acc01 = __builtin_amdgcn_wmma_f32_16x16x4_f32(false, a0, false, b1, (short)0, acc01, false, false);
            acc10 = __builtin_amdgcn_wmma_f32_16x16x4_f32(false, a1, false, b0, (short)0, acc10, false, false);
            acc11 = __builtin_amdgcn_wmma_f32_16x16x4_f32(false, a1, false, b1, (short)0, acc11, false, false);
        }

        #pragma unroll
        for (int r = 0; r < 8; ++r) {
            int rr0 = mA0 + r + 8 * half;
            int rr1 = mA1 + r + 8 * half;
            float sc0 = g3[rr0] * rsqrtf(v3[rr0] + ABN_EPS);
            float sc1 = g3[rr1] * rsqrtf(v3[rr1] + ABN_EPS);
            lds[rr0 * 33 + l16]      = leaky((acc00[r] - m3[rr0]) * sc0 + b3[rr0]);
            lds[rr0 * 33 + 16 + l16] = leaky((acc01[r] - m3[rr0]) * sc0 + b3[rr0]);
            lds[rr1 * 33 + l16]      = leaky((acc10[r] - m3[rr1]) * sc1 + b3[rr1]);
            lds[rr1 * 33 + 16 + l16] = leaky((acc11[r] - m3[rr1]) * sc1 + b3[rr1]);
        }
    }
    __syncthreads();

    // phase 2: seg (M=20 -> 2 tiles) x (Nblk=32 -> 2 tiles), waves 0..3
    if (wave < 4) {
        const int mt = wave >> 1, nt = wave & 1;
        const int m_base = mt * 16;
        const int arow = m_base + l16;
        const bool rowOK = arow < 20;
        const float* Ar = c4w + (size_t)arow * 256;
        v8f acc;
        #pragma unroll
        for (int r = 0; r < 8; ++r) acc[r] = 0.0f;
        #pragma unroll 4
        for (int k = 0; k < 256; k += 4) {
            const int kk = k + 2 * half;
            v2f a, b;
            a.x = rowOK ? Ar[kk]     : 0.0f;
            a.y = rowOK ? Ar[kk + 1] : 0.0f;
            b.x = lds[kk * 33 + nt * 16 + l16];
            b.y = lds[(kk + 1) * 33 + nt * 16 + l16];
            acc = __builtin_amdgcn_wmma_f32_16x16x4_f32(false, a, false, b, (short)0, acc, false, false);
        }
        #pragma unroll
        for (int r = 0; r < 8; ++r) {
            int rr = m_base + r + 8 * half;
            if (rr < 20)
                seg[((size_t)n * 20 + rr) * 16384 + col0 + nt * 16 + l16] = acc[r] + c4b[rr];
        }
    }
}

// ---------------------------------------------------------------------------
extern "C" void kernel_launch(void* const* d_in, const int* in_sizes, int n_in,
                              void* d_out, int out_size, void* d_ws, size_t ws_size,
                              hipStream_t stream)
{
    const float* x1   = (const float*)d_in[0];
    const float* x2   = (const float*)d_in[1];
    const float* edge = (const float*)d_in[2];
    const float* c1_w = (const float*)d_in[3];
    const float* g1   = (const float*)d_in[4];
    const float* b1   = (const float*)d_in[5];
    const float* m1   = (const float*)d_in[6];
    const float* v1   = (const float*)d_in[7];
    const float* c2_w = (const float*)d_in[8];
    const float* g2   = (const float*)d_in[9];
    const float* b2   = (const float*)d_in[10];
    const float* m2   = (const float*)d_in[11];
    const float* v2   = (const float*)d_in[12];
    const float* cs_w = (const float*)d_in[13];
    const float* cs_b = (const float*)d_in[14];
    const float* p1_w = (const float*)d_in[15];
    const float* pg   = (const float*)d_in[16];
    const float* pb   = (const float*)d_in[17];
    const float* pm   = (const float*)d_in[18];
    const float* pv   = (const float*)d_in[19];
    const float* p2_w = (const float*)d_in[20];
    const float* gw1  = (const float*)d_in[21];
    const float* gb1  = (const float*)d_in[22];
    const float* gw2  = (const float*)d_in[23];
    const float* c3_w = (const float*)d_in[24];
    const float* g3   = (const float*)d_in[25];
    const float* b3   = (const float*)d_in[26];
    const float* m3   = (const float*)d_in[27];
    const float* v3   = (const float*)d_in[28];
    const float* c4_w = (const float*)d_in[29];
    const float* c4_b = (const float*)d_in[30];

    float* seg_out  = (float*)d_out;                       // (4,20,128,128)
    float* af_out   = seg_out + (size_t)4 * 20 * 16384;    // (4,80,256)
    float* xlow_out = af_out  + (size_t)4 * 80 * 256;      // (4,20,128,128)

    char* ws = (char*)d_ws;
    float* t1   = (float*)(ws + 0);               // (4,256,4096)   16.78 MB
    float* ub   = (float*)(ws + 16777216ull);     // (4,256,4096)   16.78 MB
    float* x2p  = (float*)(ws + 33554432ull);     // (4,48,16384)   12.58 MB (also t_p1)
    float* xb   = (float*)(ws + 46137344ull);     // (4,256,16384)  67.1 MB  (x -> xf)
    float* proj = (float*)(ws + 0);               // (4,80,16384)   reuses t1+u (both dead)
    float* eb   = (float*)(ws + 113246208ull);    // (4,16384)
    int*   idxb = (int*)  (ws + 113508352ull);    // (4,80)
    float* hgb  = (float*)(ws + 113509888ull);    // (4,256,80)
    float* xnb  = (float*)(ws + 113837568ull);    // (4,256,80)   -> total ~109 MiB

    dim3 blk(128);

    // e = sigmoid(e1-e0) on nearest-upsampled edge
    edge_softmax_kernel<<<256, 256, 0, stream>>>(edge, eb);

    // t1 = abn1(c1_w @ x1)                M=256 K=512 N=4096
    gemm_wmma_f32<EPI_ABN, false><<<dim3(32, 8, 4), blk, 0, stream>>>(
        c1_w, 0, 512, x1, (size_t)512 * 4096, t1, (size_t)256 * 4096,
        256, 4096, 512, g1, b1, m1, v1, 0);

    // u = cs_w[:, :256] @ t1              M=256 K=256 N=4096 (lda=304)
    gemm_wmma_f32<EPI_NONE, false><<<dim3(32, 8, 4), blk, 0, stream>>>(
        cs_w, 0, 304, t1, (size_t)256 * 4096, ub, (size_t)256 * 4096,
        256, 4096, 256, nullptr, nullptr, nullptr, nullptr, 0);

    // x = bilinear(u) + cs_b
    upsample_bias_kernel<<<65536, 256, 0, stream>>>(ub, cs_b, xb);

    // x2p = abn2(c2_w @ x2)               M=48 K=256 N=16384
    gemm_wmma_f32<EPI_ABN, false><<<dim3(128, 2, 4), blk, 0, stream>>>(
        c2_w, 0, 256, x2, (size_t)256 * 16384, x2p, (size_t)48 * 16384,
        48, 16384, 256, g2, b2, m2, v2, 0);

    // x += cs_w[:, 256:304] @ x2p         M=256 K=48 N=16384 (accumulate)
    gemm_wmma_f32<EPI_NONE, true><<<dim3(128, 8, 4), blk, 0, stream>>>(
        cs_w + 256, 0, 304, x2p, (size_t)48 * 16384, xb, (size_t)256 * 16384,
        256, 16384, 48, nullptr, nullptr, nullptr, nullptr, 0);

    // t_p1 = abn_p(p1_w @ x)  (into x2p buffer)   M=48 K=256 N=16384
    gemm_wmma_f32<EPI_ABN, false><<<dim3(128, 2, 4), blk, 0, stream>>>(
        p1_w, 0, 256, xb, (size_t)256 * 16384, x2p, (size_t)48 * 16384,
        48, 16384, 256, pg, pb, pm, pv, 0);

    // x_low = p2_w @ t_p1                 M=20 K=48 N=16384
    gemm_wmma_f32<EPI_NONE, false><<<dim3(128, 1, 4), blk, 0, stream>>>(
        p2_w, 0, 48, x2p, (size_t)48 * 16384, xlow_out, (size_t)20 * 16384,
        20, 16384, 48, nullptr, nullptr, nullptr, nullptr, 0);

    // top-4 indices per (n, c)
    topk4_kernel<<<80, 256, 0, stream>>>(xlow_out, idxb);

    // anchor_feature gather (folds 1-e)
    gather_anchors_kernel<<<320, 256, 0, stream>>>(xb, eb, idxb, af_out);

    // proj = (af @ x) * e                 M=80 K=256 N=16384, col-scale epi
    gemm_wmma_f32<EPI_COLSCALE, false><<<dim3(128, 3, 4), blk, 0, stream>>>(
        af_out, (size_t)80 * 256, 256, xb, (size_t)256 * 16384, proj, (size_t)80 * 16384,
        80, 16384, 256, eb, nullptr, nullptr, nullptr, 16384);

    // hg = gw1 @ x_anchor + gb1 - x_anchor
    hg_kernel<<<320, 256, 0, stream>>>(gw1, gb1, af_out, hgb);

    // xn = relu(gw2 @ hg)                 M=256 K=256 N=80
    gemm_wmma_f32<EPI_RELU, false><<<dim3(1, 8, 4), blk, 0, stream>>>(
        gw2, 0, 256, hgb, (size_t)256 * 80, xnb, (size_t)256 * 80,
        256, 80, 256, nullptr, nullptr, nullptr, nullptr, 0);

    // proj = softmax_k(proj) * e
    softmax_proj_kernel<<<256, 256, 0, stream>>>(proj, eb);

    // xf = x + xn @ proj (in place)       M=256 K=80 N=16384 (accumulate)
    gemm_wmma_f32<EPI_NONE, true><<<dim3(128, 8, 4), blk, 0, stream>>>(
        xnb, (size_t)256 * 80, 80, proj, (size_t)80 * 16384, xb, (size_t)256 * 16384,
        256, 16384, 80, nullptr, nullptr, nullptr, nullptr, 0);

    // seg = c4_w @ abn3(c3_w @ xf) + c4_b (fused via LDS)
    fused_c3abn_c4_kernel<<<dim3(512, 4), 256, 0, stream>>>(
        xb, c3_w, g3, b3, m3, v3, c4_w, c4_b, seg_out);

    (void)in_sizes; (void)n_in; (void)out_size; (void)ws_size;
}